// GAT_27693949125042
// MI455X (gfx1250) — compile-verified
//
#include <hip/hip_runtime.h>
#include <math.h>

typedef __attribute__((ext_vector_type(2))) float v2f;
typedef __attribute__((ext_vector_type(8))) float v8f;

#define NEG_SLOPE 0.2f
#define BN_EPS 1e-5f
#define HID 64
#define NGRAPH 512

// ---------- float atomic max via sign-split integer atomics (init must be -inf)
__device__ __forceinline__ void atomicMaxF32(float* addr, float v) {
  if (v >= 0.0f) atomicMax((int*)addr, __float_as_int(v));
  else           atomicMin((unsigned int*)addr, __float_as_uint(v));
}

// ---------- dual GEMM: Xl = A@Wl + bl, Xr = A@Wr + br   (A: [nrows,64], W: [64,64])
// one wave per 16-row tile; V_WMMA_F32_16X16X4_F32, K=64 in 16 steps
__global__ __launch_bounds__(256)
void dual_gemm64(const float* __restrict__ A,
                 const float* __restrict__ Wl, const float* __restrict__ Wr,
                 const float* __restrict__ bl, const float* __restrict__ br,
                 float* __restrict__ Xl, float* __restrict__ Xr, int nrows) {
  int wave = (blockIdx.x * blockDim.x + threadIdx.x) >> 5;
  int lane = threadIdx.x & 31;
  int half = lane >> 4;      // 0: lanes 0-15, 1: lanes 16-31
  int lr   = lane & 15;
  int m0 = wave * 16;
  if (m0 >= nrows) return;   // wave-uniform -> EXEC stays all-ones for WMMA

  const int ar = m0 + lr;    // A row handled by this lane (16x4 A layout)
  #pragma unroll
  for (int nt = 0; nt < 4; ++nt) {
    const int n0 = nt * 16;
    v8f cl = {}; v8f cr = {};
    #pragma unroll
    for (int k0 = 0; k0 < 64; k0 += 4) {
      const int ka = k0 + 2 * half;
      v2f a;  a.x = A[ar * 64 + ka];            a.y = A[ar * 64 + ka + 1];
      v2f bL; bL.x = Wl[ka * 64 + n0 + lr];     bL.y = Wl[(ka + 1) * 64 + n0 + lr];
      v2f bR; bR.x = Wr[ka * 64 + n0 + lr];     bR.y = Wr[(ka + 1) * 64 + n0 + lr];
      cl = __builtin_amdgcn_wmma_f32_16x16x4_f32(false, a, false, bL, (short)0, cl, false, false);
      cr = __builtin_amdgcn_wmma_f32_16x16x4_f32(false, a, false, bR, (short)0, cr, false, false);
    }
    const float addl = bl[n0 + lr];
    const float addr_ = br[n0 + lr];
    #pragma unroll
    for (int v = 0; v < 8; ++v) {
      const int row = m0 + v + 8 * half;        // C/D layout: VGPR v = rows v / v+8
      Xl[row * 64 + n0 + lr] = cl[v] + addl;
      Xr[row * 64 + n0 + lr] = cr[v] + addr_;
    }
  }
}

// ---------- generic GEMM (K=64): C = A@W + b, ncols multiple of 16
__global__ __launch_bounds__(256)
void gemm64_wmma(const float* __restrict__ A, const float* __restrict__ W,
                 const float* __restrict__ bias, float* __restrict__ C,
                 int nrows, int ncols) {
  int wave = (blockIdx.x * blockDim.x + threadIdx.x) >> 5;
  int lane = threadIdx.x & 31;
  int half = lane >> 4, lr = lane & 15;
  int ntiles = ncols >> 4;
  int tm = wave / ntiles, tn = wave % ntiles;
  int m0 = tm * 16, n0 = tn * 16;
  if (m0 >= nrows) return;
  const int ar = m0 + lr;
  v8f c = {};
  #pragma unroll
  for (int k0 = 0; k0 < 64; k0 += 4) {
    const int ka = k0 + 2 * half;
    v2f a; a.x = A[ar * 64 + ka];              a.y = A[ar * 64 + ka + 1];
    v2f b; b.x = W[ka * ncols + n0 + lr];      b.y = W[(ka + 1) * ncols + n0 + lr];
    c = __builtin_amdgcn_wmma_f32_16x16x4_f32(false, a, false, b, (short)0, c, false, false);
  }
  const float bb = bias[n0 + lr];
  #pragma unroll
  for (int v = 0; v < 8; ++v) {
    const int row = m0 + v + 8 * half;
    C[row * ncols + n0 + lr] = c[v] + bb;
  }
}

// ---------- per-layer init: m=-inf, s=0, out[n,c]=bias[c]
__global__ __launch_bounds__(256)
void prep_layer(float* __restrict__ m, float* __restrict__ s,
                float* __restrict__ out, const float* __restrict__ bias, int N) {
  long long t = (long long)blockIdx.x * blockDim.x + threadIdx.x;
  if (t < (long long)N * 64) out[t] = bias[t & 63];
  if (t < N) { m[t] = -__builtin_inff(); s[t] = 0.f; }
}

// ---------- edge pass 1: e = att . leakyrelu(xl[src]+xr[dst]); per-dst atomic max
__global__ __launch_bounds__(256)
void edge_logits(const int* __restrict__ ei, int E, int N,
                 const float* __restrict__ xl, const float* __restrict__ xr,
                 const float* __restrict__ att, float* __restrict__ ebuf,
                 float* __restrict__ m) {
  long long t = (long long)blockIdx.x * blockDim.x + threadIdx.x;
  int edge = (int)(t >> 4);
  int j = (int)(t & 15);
  const int ET = E + N;
  if (edge >= ET) return;
  int src, dst;
  if (edge < E) { src = ei[edge]; dst = ei[E + edge]; }
  else          { src = dst = edge - E; }

  float4 a = ((const float4*)att)[j];
  float4 u = ((const float4*)(xl + (long long)src * 64))[j];
  float4 v = ((const float4*)(xr + (long long)dst * 64))[j];
  float h0 = u.x + v.x, h1 = u.y + v.y, h2 = u.z + v.z, h3 = u.w + v.w;
  h0 = h0 > 0.f ? h0 : NEG_SLOPE * h0;
  h1 = h1 > 0.f ? h1 : NEG_SLOPE * h1;
  h2 = h2 > 0.f ? h2 : NEG_SLOPE * h2;
  h3 = h3 > 0.f ? h3 : NEG_SLOPE * h3;
  float acc = a.x * h0 + a.y * h1 + a.z * h2 + a.w * h3;
  acc += __shfl_xor(acc, 8, 32);   // reduction stays inside the 16-lane team
  acc += __shfl_xor(acc, 4, 32);
  acc += __shfl_xor(acc, 2, 32);
  acc += __shfl_xor(acc, 1, 32);
  if (j == 0) { ebuf[edge] = acc; atomicMaxF32(&m[dst], acc); }
}

// ---------- edge pass 2: w = exp(e - m[dst]); s[dst] += w
__global__ __launch_bounds__(256)
void edge_expsum(const int* __restrict__ ei, int E, int N,
                 float* __restrict__ ebuf, const float* __restrict__ m,
                 float* __restrict__ s) {
  int edge = blockIdx.x * blockDim.x + threadIdx.x;
  const int ET = E + N;
  if (edge >= ET) return;
  int dst = (edge < E) ? ei[E + edge] : (edge - E);
  float w = __expf(ebuf[edge] - m[dst]);
  ebuf[edge] = w;
  atomicAdd(&s[dst], w);
}

// ---------- edge pass 3: out[dst] += (w/s[dst]) * xl[src]
__global__ __launch_bounds__(256)
void edge_scatter(const int* __restrict__ ei, int E, int N,
                  const float* __restrict__ xl, const float* __restrict__ ebuf,
                  const float* __restrict__ s, float* __restrict__ out) {
  long long t = (long long)blockIdx.x * blockDim.x + threadIdx.x;
  int edge = (int)(t >> 4);
  int j = (int)(t & 15);
  const int ET = E + N;
  if (edge >= ET) return;
  int src, dst;
  if (edge < E) { src = ei[edge]; dst = ei[E + edge]; }
  else          { src = dst = edge - E; }
  float alpha = ebuf[edge] / s[dst];
  float4 hl = ((const float4*)(xl + (long long)src * 64))[j];
  float* o = out + (long long)dst * 64 + j * 4;
  atomicAdd(o + 0, alpha * hl.x);
  atomicAdd(o + 1, alpha * hl.y);
  atomicAdd(o + 2, alpha * hl.z);
  atomicAdd(o + 3, alpha * hl.w);
}

__global__ __launch_bounds__(256)
void relu_inplace(float* __restrict__ h, long long n) {
  long long t = (long long)blockIdx.x * blockDim.x + threadIdx.x;
  if (t < n) h[t] = fmaxf(h[t], 0.f);
}

__global__ __launch_bounds__(256)
void pool_init(float* __restrict__ pooled, float* __restrict__ cnts) {
  int t = blockIdx.x * blockDim.x + threadIdx.x;
  if (t < NGRAPH * HID) pooled[t] = 0.f;
  if (t < NGRAPH) cnts[t] = 0.f;
}

__global__ __launch_bounds__(256)
void pool_acc(const float* __restrict__ h, const int* __restrict__ batch,
              float* __restrict__ pooled, float* __restrict__ cnts, int N) {
  long long t = (long long)blockIdx.x * blockDim.x + threadIdx.x;
  if (t >= (long long)N * 64) return;
  int node = (int)(t >> 6), c = (int)(t & 63);
  int g = batch[node];
  atomicAdd(&pooled[g * 64 + c], h[t]);
  if (c == 0) atomicAdd(&cnts[g], 1.0f);
}

__global__ __launch_bounds__(256)
void pool_div(float* __restrict__ pooled, const float* __restrict__ cnts) {
  int t = blockIdx.x * blockDim.x + threadIdx.x;
  if (t < NGRAPH * HID) pooled[t] /= cnts[t >> 6];
}

// batch-norm stats over axis 0 of z[512,256]: stats[0:256]=mu, stats[256:512]=var
__global__ __launch_bounds__(256)
void bn_stats(const float* __restrict__ z, float* __restrict__ stats) {
  int c = threadIdx.x;
  float sum = 0.f, sq = 0.f;
  for (int g = 0; g < NGRAPH; ++g) {
    float v = z[g * 256 + c];
    sum += v; sq += v * v;
  }
  float mu = sum * (1.0f / NGRAPH);
  float var = sq * (1.0f / NGRAPH) - mu * mu;
  stats[c] = mu;
  stats[256 + c] = var;
}

// normalize + relu + final [256x2] GEMM
__global__ __launch_bounds__(256)
void head_kernel(const float* __restrict__ z, const float* __restrict__ stats,
                 const float* __restrict__ W2, const float* __restrict__ b2,
                 float* __restrict__ out) {
  int t = blockIdx.x * blockDim.x + threadIdx.x;
  if (t >= NGRAPH * 2) return;
  int g = t >> 1, o = t & 1;
  float acc = 0.f;
  for (int c = 0; c < 256; ++c) {
    float zn = (z[g * 256 + c] - stats[c]) * rsqrtf(stats[256 + c] + BN_EPS);
    zn = fmaxf(zn, 0.f);
    acc += zn * W2[c * 2 + o];
  }
  out[t] = acc + b2[o];
}

static inline int cdiv(long long a, long long b) { return (int)((a + b - 1) / b); }

extern "C" void kernel_launch(void* const* d_in, const int* in_sizes, int n_in,
                              void* d_out, int out_size, void* d_ws, size_t ws_size,
                              hipStream_t stream) {
  const float* x   = (const float*)d_in[0];
  const int*   ei  = (const int*)d_in[1];     // [2, E] flattened
  const int*   bat = (const int*)d_in[3];
  const int N = in_sizes[0] / 64;
  const int E = in_sizes[2];                  // edge_weight count == E
  const int ET = E + N;

  // params: per conv layer {Wl, Wr, bl, br, att, bias}
  const float* P[32];
  for (int i = 4; i < n_in && i < 36; ++i) P[i - 4] = (const float*)d_in[i];
  const float* linW  = P[18];  // d_in[22]
  const float* linB  = P[19];
  const float* lin2W = P[20];
  const float* lin2B = P[21];

  // workspace carve-up (floats)
  const long long NH = (long long)N * 64;
  float* ws     = (float*)d_ws;
  float* hA     = ws;
  float* hB     = hA + NH;
  float* xl     = hB + NH;
  float* xr     = xl + NH;
  float* mx     = xr + NH;
  float* sm     = mx + N;
  float* eb     = sm + N;
  float* pooled = eb + ET;
  float* cnts   = pooled + NGRAPH * HID;
  float* z      = cnts + NGRAPH;
  float* stats  = z + NGRAPH * 256;
  (void)ws_size;

  const int waves = N / 16;                       // 100000/16 = 6250 exact
  const int gemmBlocks = cdiv((long long)waves * 32, 256);
  const int nodeElemBlocks = cdiv(NH, 256);
  const int edgeTeamBlocks = cdiv((long long)ET * 16, 256);
  const int edgeBlocks = cdiv(ET, 256);

  const float* in = x;
  for (int layer = 0; layer < 3; ++layer) {
    const float* Wl = P[layer * 6 + 0];
    const float* Wr = P[layer * 6 + 1];
    const float* bl = P[layer * 6 + 2];
    const float* br = P[layer * 6 + 3];
    const float* at = P[layer * 6 + 4];
    const float* bi = P[layer * 6 + 5];
    float* out = (layer == 1) ? hB : hA;          // L1->hA, L2->hB, L3->hA

    dual_gemm64<<<gemmBlocks, 256, 0, stream>>>(in, Wl, Wr, bl, br, xl, xr, N);
    prep_layer<<<nodeElemBlocks, 256, 0, stream>>>(mx, sm, out, bi, N);
    edge_logits<<<edgeTeamBlocks, 256, 0, stream>>>(ei, E, N, xl, xr, at, eb, mx);
    edge_expsum<<<edgeBlocks, 256, 0, stream>>>(ei, E, N, eb, mx, sm);
    edge_scatter<<<edgeTeamBlocks, 256, 0, stream>>>(ei, E, N, xl, eb, sm, out);
    if (layer < 2) relu_inplace<<<nodeElemBlocks, 256, 0, stream>>>(out, NH);
    in = out;
  }

  // global mean pool (final features in hA)
  pool_init<<<cdiv(NGRAPH * HID, 256), 256, 0, stream>>>(pooled, cnts);
  pool_acc<<<nodeElemBlocks, 256, 0, stream>>>(hA, bat, pooled, cnts, N);
  pool_div<<<cdiv(NGRAPH * HID, 256), 256, 0, stream>>>(pooled, cnts);

  // z = pooled @ linW + linB   (512x64 @ 64x256) via WMMA
  const int linWaves = (NGRAPH / 16) * (256 / 16);   // 512 waves
  gemm64_wmma<<<cdiv((long long)linWaves * 32, 256), 256, 0, stream>>>(
      pooled, linW, linB, z, NGRAPH, 256);

  bn_stats<<<1, 256, 0, stream>>>(z, stats);
  head_kernel<<<cdiv(NGRAPH * 2, 256), 256, 0, stream>>>(z, stats, lin2W, lin2B,
                                                         (float*)d_out);
}